// AdaptiveHadamardTransform_10419590660571
// MI455X (gfx1250) — compile-verified
//
#include <hip/hip_runtime.h>

// CDNA5 / gfx1250, wave32.
// 4096-pt Walsh-Hadamard = H16 (x) H16 (x) H16, each axis done as 16x16x16
// matmuls with the +/-1 Hadamard matrix using V_WMMA_F32_16X16X4_F32
// (exact f32: +/-1 products are exact, accumulation is f32 adds).

typedef __attribute__((ext_vector_type(2))) float v2f;
typedef __attribute__((ext_vector_type(8))) float v8f;

// Padded LDS strides (dwords): S1 for the j axis, S2 for the i axis.
// S1=20 -> 16 distinct banks for stride-S1 access patterns, 16B-aligned rows.
#define S1 20
#define S2 320   // 16 * S1
#define NROW_ELEMS 4096

// H16[r][c] = (-1)^popcount(r & c), built as a float sign flip (no memory).
__device__ __forceinline__ float hadsign(int r, int c) {
    unsigned s = (unsigned)(__popc(r & c) & 1) << 31;
    return __int_as_float(0x3F800000u | s);
}

__global__ __launch_bounds__(512)
void fht4096_wmma_kernel(const float* __restrict__ x,
                         const float* __restrict__ scale,
                         const float* __restrict__ shift,
                         float* __restrict__ out)
{
    __shared__ float lds[16 * S2];   // 20 KB

    const int tid  = threadIdx.x;
    const int lane = tid & 31;
    const int wave = tid >> 5;       // 0..15
    const int half = lane >> 4;      // 0 or 1 (half-wave)
    const int l15  = lane & 15;

    const size_t rowbase = (size_t)blockIdx.x * NROW_ELEMS;

    // ---- load one row (16 KB) into padded LDS layout: addr(i,j,k)=i*S2+j*S1+k
    {
        const float4* xr = (const float4*)(x + rowbase);
        #pragma unroll
        for (int it = 0; it < 2; ++it) {
            int v = tid + it * 512;          // float4 index 0..1023
            int g = v << 2;                  // element index 0..4095
            float4 d = xr[v];
            int i = g >> 8, j = (g >> 4) & 15, k = g & 15;  // k in {0,4,8,12}
            *(float4*)&lds[i * S2 + j * S1 + k] = d;
        }
    }
    __syncthreads();

    // ---- stage 1: Y_i = X_i * H   (transform along k); wave w owns i = w
    {
        const int base = wave * S2;
        v8f acc = {};
        #pragma unroll
        for (int q = 0; q < 4; ++q) {
            int kb = 4 * q + 2 * half;       // per-half-wave K pair
            v2f a, b;
            // A = X_i : A[m][kb..kb+1], m = l15  (16x4 layout, 2 VGPRs)
            a.x = lds[base + l15 * S1 + kb];
            a.y = lds[base + l15 * S1 + kb + 1];
            // B = H   : B[kb..kb+1][n], n = l15 (4x16 layout, 2 VGPRs)
            b.x = hadsign(kb,     l15);
            b.y = hadsign(kb + 1, l15);
            acc = __builtin_amdgcn_wmma_f32_16x16x4_f32(false, a, false, b,
                                                        (short)0, acc, false, false);
        }
        // D 16x16 f32: VGPR r -> row (r + 8*half), col l15
        #pragma unroll
        for (int r = 0; r < 8; ++r)
            lds[base + (r + 8 * half) * S1 + l15] = acc[r];
    }
    // stage 1 -> stage 2 is within one wave's own LDS block; per-wave LDS ops
    // execute in order, so no workgroup barrier is needed here.

    // ---- stage 2: Z_i = H * Y_i   (transform along j); same i block
    {
        const int base = wave * S2;
        v8f acc = {};
        #pragma unroll
        for (int q = 0; q < 4; ++q) {
            int kb = 4 * q + 2 * half;
            v2f a, b;
            a.x = hadsign(l15, kb);          // A = H : A[m][kb..kb+1]
            a.y = hadsign(l15, kb + 1);
            b.x = lds[base + kb       * S1 + l15];   // B = Y : rows kb..kb+1
            b.y = lds[base + (kb + 1) * S1 + l15];
            acc = __builtin_amdgcn_wmma_f32_16x16x4_f32(false, a, false, b,
                                                        (short)0, acc, false, false);
        }
        #pragma unroll
        for (int r = 0; r < 8; ++r)
            lds[base + (r + 8 * half) * S1 + l15] = acc[r];
    }
    __syncthreads();   // stage 3 reads across all i blocks

    // ---- stage 3: W[.,j,k] = H * A_k, A_k[i][j] = Z[i][j][k]; wave w owns k=w
    {
        const int k = wave;
        v8f acc = {};
        #pragma unroll
        for (int q = 0; q < 4; ++q) {
            int kb = 4 * q + 2 * half;
            v2f a, b;
            a.x = hadsign(l15, kb);          // A = H
            a.y = hadsign(l15, kb + 1);
            b.x = lds[kb       * S2 + l15 * S1 + k];  // B[i=kb..][j=l15]
            b.y = lds[(kb + 1) * S2 + l15 * S1 + k];
            acc = __builtin_amdgcn_wmma_f32_16x16x4_f32(false, a, false, b,
                                                        (short)0, acc, false, false);
        }
        #pragma unroll
        for (int r = 0; r < 8; ++r)
            lds[(r + 8 * half) * S2 + l15 * S1 + k] = acc[r];
    }
    __syncthreads();

    // ---- epilogue: out = scale * (W / 64) + shift   (1/sqrt(4096) == 1/64)
    {
        const float inv = 1.0f / 64.0f;
        float4* orow = (float4*)(out + rowbase);
        const float4* sc4 = (const float4*)scale;
        const float4* sh4 = (const float4*)shift;
        #pragma unroll
        for (int it = 0; it < 2; ++it) {
            int v = tid + it * 512;
            int g = v << 2;
            int i = g >> 8, j = (g >> 4) & 15, k = g & 15;
            float4 w = *(const float4*)&lds[i * S2 + j * S1 + k];
            float4 s = sc4[v];
            float4 b = sh4[v];
            float4 o;
            o.x = fmaf(s.x * inv, w.x, b.x);
            o.y = fmaf(s.y * inv, w.y, b.y);
            o.z = fmaf(s.z * inv, w.z, b.z);
            o.w = fmaf(s.w * inv, w.w, b.w);
            orow[v] = o;
        }
    }
}

extern "C" void kernel_launch(void* const* d_in, const int* in_sizes, int n_in,
                              void* d_out, int out_size, void* d_ws, size_t ws_size,
                              hipStream_t stream) {
    (void)in_sizes; (void)n_in; (void)d_ws; (void)ws_size;
    const float* x     = (const float*)d_in[0];
    const float* scale = (const float*)d_in[1];
    const float* shift = (const float*)d_in[2];
    float* out = (float*)d_out;

    const int rows = out_size / NROW_ELEMS;   // 4 * 4096 = 16384
    fht4096_wmma_kernel<<<rows, 512, 0, stream>>>(x, scale, shift, out);
}